// RNNLayer_76141180223840
// MI455X (gfx1250) — compile-verified
//
#include <hip/hip_runtime.h>

// Low-rank RNN step for MI455X (gfx1250, wave32).
//   Y      = tanh(h) @ V^T                      (pass 1, V_WMMA_F32_16X16X4_F32)
//   out    = h + 0.01 * ( (Y @ U^T)/N + ff - h) (pass 2, coalesced float4)
// HBM-bound: ~128 MB compulsory traffic -> ~5.5 us floor at 23.3 TB/s.

typedef __attribute__((ext_vector_type(2))) float v2f;
typedef __attribute__((ext_vector_type(4))) float v4f;
typedef __attribute__((ext_vector_type(8))) float v8f;

#define N_HID   8192
#define RANK_R  8
#define BATCH_B 2048
#define KSPLIT  16
#define KCHUNK  (N_HID / KSPLIT)   // 512
#define DT_TAU  0.01f              // TIME_STEP_SIZE / TAU
#define INV_N   (1.0f / 8192.0f)

// Branchless tanh on the TRANS pipe: tanh(x) = 1 - 2/(exp2(2*log2e*x)+1).
// v_mul + v_exp_f32 + v_add + v_rcp_f32 + v_fma; saturates to +/-1 via
// inf/0 without branches, NaN propagates. EXEC stays all-1s (WMMA-safe).
__device__ __forceinline__ float fast_tanh(float x) {
  float e = __builtin_amdgcn_exp2f(x * 2.885390081777927f); // exp(2x)
  float r = __builtin_amdgcn_rcpf(e + 1.0f);
  return __builtin_fmaf(-2.0f, r, 1.0f);
}

// ---------------------------------------------------------------------------
// Pass 1: Ypart[split][b][r] = sum_{k in chunk} tanh(h[b,k]) * V[r,k]
// One wave32 per (16-row batch tile, 512-wide K chunk); 128*16 = 2048 waves.
//
// A-operand (16x4 f32): lane L holds row M=L%16; VGPR0/1 hold K = 2*(L/16),
// 2*(L/16)+1  -> each lane does one 8-byte sequential load per WMMA, so the
// 16 rows stream linearly and cachelines are fully consumed across the loop.
// B-operand (4x16 f32): lane L holds column col=L%16 of the V^T chunk;
// columns 8..15 duplicate 0..7 (harmless: WMMA columns don't mix, never
// stored).
// ---------------------------------------------------------------------------
__global__ __launch_bounds__(256) void rnn_rank_proj(
    const float* __restrict__ h, const float* __restrict__ V,
    float* __restrict__ Ypart)
{
  const int lane  = threadIdx.x & 31;
  const int wave  = threadIdx.x >> 5;
  const int wgid  = blockIdx.x * 8 + wave;      // 0..2047
  const int tile  = wgid / KSPLIT;              // 0..127 (16 batch rows each)
  const int split = wgid % KSPLIT;              // 0..15  (512-wide K chunk)

  const int mrow  = lane & 15;                  // A row for this lane
  const int khalf = (lane >> 4) << 1;           // K sub-offset: 0 or 2
  const int col   = lane & 15;                  // B column (rank idx, valid <8)

  const float* hrow = h + (size_t)(tile * 16 + mrow) * N_HID
                        + (size_t)split * KCHUNK + khalf;
  const float* vrow = V + (size_t)(col & 7) * N_HID
                        + (size_t)split * KCHUNK + khalf;

  v8f acc = {};
  #pragma unroll 8
  for (int k = 0; k < KCHUNK; k += 4) {
    v2f ha = *(const v2f*)(hrow + k);           // 8B sequential per lane
    v2f a;
    a.x = fast_tanh(ha.x);
    a.y = fast_tanh(ha.y);
    v2f b = *(const v2f*)(vrow + k);            // V^T chunk, L2-resident
    // D = A(16x4,f32) * B(4x16,f32) + C   ->  v_wmma_f32_16x16x4_f32
    acc = __builtin_amdgcn_wmma_f32_16x16x4_f32(
        /*neg_a=*/false, a, /*neg_b=*/false, b,
        /*c_mod=*/(short)0, acc, /*reuse_a=*/false, /*reuse_b=*/false);
  }

  // C/D layout: lane L, VGPR i  ->  D[M = i + 8*(L/16)][Ncol = L%16]
  if (col < RANK_R) {
    const int mbase = tile * 16 + ((lane >> 4) << 3);
    float* yout = Ypart + ((size_t)split * BATCH_B + mbase) * RANK_R + col;
    #pragma unroll
    for (int i = 0; i < 8; ++i)
      yout[(size_t)i * RANK_R] = acc[i];
  }
}

// ---------------------------------------------------------------------------
// Pass 2: out[b,n] = h[b,n] + 0.01*( dot(Y[b,:],U[n,:])/N + ff[n] - h[b,n] )
// One block per (batch row x 1024 columns); float4 coalesced h/ff/out
// traffic; U (256 KB), ff and Ypart (1 MB) are L2-resident by pass 1.
// ---------------------------------------------------------------------------
__global__ __launch_bounds__(256) void rnn_update(
    const float* __restrict__ h, const float* __restrict__ ff,
    const float* __restrict__ U, const float* __restrict__ Ypart,
    float* __restrict__ out)
{
  __shared__ float Yb[RANK_R];
  const int b = blockIdx.y;

  if (threadIdx.x < RANK_R) {                   // reduce K-split partials once
    float s = 0.f;
    #pragma unroll
    for (int sp = 0; sp < KSPLIT; ++sp)
      s += Ypart[((size_t)sp * BATCH_B + b) * RANK_R + threadIdx.x];
    Yb[threadIdx.x] = s * INV_N;                // fold the 1/N of (U@V)/N here
  }
  __syncthreads();

  float y[RANK_R];
  #pragma unroll
  for (int r = 0; r < RANK_R; ++r) y[r] = Yb[r];

  const int n0 = (blockIdx.x * 256 + threadIdx.x) * 4;
  v4f hv = *(const v4f*)(h  + (size_t)b * N_HID + n0);
  v4f fv = *(const v4f*)(ff + n0);

  v4f z;
  #pragma unroll
  for (int j = 0; j < 4; ++j) {
    const float* urow = U + (size_t)(n0 + j) * RANK_R;
    v4f u0 = *(const v4f*)(urow);
    v4f u1 = *(const v4f*)(urow + 4);
    z[j] = u0.x*y[0] + u0.y*y[1] + u0.z*y[2] + u0.w*y[3]
         + u1.x*y[4] + u1.y*y[5] + u1.z*y[6] + u1.w*y[7];
  }

  v4f hnew;
  #pragma unroll
  for (int j = 0; j < 4; ++j)
    hnew[j] = hv[j] + DT_TAU * (z[j] + fv[j] - hv[j]);
  *(v4f*)(out + (size_t)b * N_HID + n0) = hnew;
}

// ---------------------------------------------------------------------------
// Inputs (setup_inputs order): 0=feedforward_input[8192], 1=hidden_state
// [2048*8192], 2=U[8192*8], 3=V[8*8192].  Output: float32 [2048*8192].
// Workspace: KSPLIT*BATCH*RANK*4 = 1 MB of d_ws for Y partials (disjoint
// writes, no atomics -> bitwise deterministic).
// ---------------------------------------------------------------------------
extern "C" void kernel_launch(void* const* d_in, const int* in_sizes, int n_in,
                              void* d_out, int out_size, void* d_ws, size_t ws_size,
                              hipStream_t stream) {
  const float* ff = (const float*)d_in[0];
  const float* h  = (const float*)d_in[1];
  const float* U  = (const float*)d_in[2];
  const float* V  = (const float*)d_in[3];
  float* out   = (float*)d_out;
  float* Ypart = (float*)d_ws;   // needs 1 MB

  // Pass 1: 128 tiles * 16 K-splits = 2048 waves -> 256 blocks of 8 waves.
  rnn_rank_proj<<<256, 256, 0, stream>>>(h, V, Ypart);

  // Pass 2: grid (8192/1024 cols, 2048 rows).
  dim3 g2(N_HID / 1024, BATCH_B);
  rnn_update<<<g2, 256, 0, stream>>>(h, ff, U, Ypart, out);
}